// Model_76544907149549
// MI455X (gfx1250) — compile-verified
//
#include <hip/hip_runtime.h>
#include <math.h>

typedef __attribute__((ext_vector_type(2))) float v2f;
typedef __attribute__((ext_vector_type(8))) float v8f;

#define NB 8
#define HEADS 4
#define FFD 1024

__device__ __forceinline__ float gelu_f(float x) {
  return 0.5f * x * (1.0f + erff(x * 0.70710678118654752440f));
}

__device__ __forceinline__ v8f wmma4(v2f a, v2f b, v8f c) {
  return __builtin_amdgcn_wmma_f32_16x16x4_f32(false, a, false, b, (short)0, c,
                                               false, false);
}

// ---------------------------------------------------------------------------
// FAST batched GEMM on the CDNA5 fp32 matrix pipe (V_WMMA_F32_16X16X4_F32).
// Preconditions (enforced by host dispatch): M%16==0, N%16==0, K%4==0, K>=8.
// C[M,N] = act(alpha * A[M,K] @ B + bias[N]) + residual   (batched over z)
// B is [K,N] row-major (transB=0) or [N,K] row-major (transB=1).
// Each wave computes a 16(M) x 32(N) strip: one A fragment feeds two
// accumulators; loads unpredicated, 2-stage software pipeline; stores
// unguarded. Fragment layout per ISA 7.12.2:
//   A 16x4 : lane r=lane&15 -> row m0+r ; vgpr v, half=lane>>4 -> K = 2*half+v
//   B 4x16 : lane r -> col n0+r        ; vgpr v, half        -> K = 2*half+v
//   C 16x16: lane r -> col n0+r        ; vgpr v, half        -> M = m0+v+8*half
// ---------------------------------------------------------------------------
__global__ void __launch_bounds__(128)
gemm_f32_wmma_fast(const float* __restrict__ A, const float* __restrict__ Bm,
                   const float* __restrict__ bias, const float* __restrict__ Rm,
                   float* __restrict__ C,
                   int M, int N, int K, int lda, int ldb, int ldc,
                   long sA, long sB, long sC, long sR,
                   int transB, int act, float alpha)
{
  const int wave = threadIdx.x >> 5;
  const int lane = threadIdx.x & 31;
  const int half = lane >> 4;
  const int r    = lane & 15;
  const int m0 = blockIdx.y * 16;
  const int n0 = blockIdx.x * 128 + wave * 32;
  if (n0 >= N) return;                       // wave-uniform
  const bool sub1 = (n0 + 32) <= N;          // wave-uniform (N%16==0)
  const long bz = blockIdx.z;
  const float* Ab = A + bz * sA;
  const float* Bb = Bm + bz * sB;
  const int row  = m0 + r;
  const int col0 = n0 + r;

  v8f acc0 = {0.f, 0.f, 0.f, 0.f, 0.f, 0.f, 0.f, 0.f};
  v8f acc1 = {0.f, 0.f, 0.f, 0.f, 0.f, 0.f, 0.f, 0.f};

  const float* ap = Ab + (long)row * lda + 2 * half;
  if (sub1) {
    if (transB) {
      const float* bp0 = Bb + (long)col0 * ldb + 2 * half;
      const float* bp1 = bp0 + 16L * ldb;
      v2f a_c = *(const v2f*)ap;
      v2f b0_c = *(const v2f*)bp0;
      v2f b1_c = *(const v2f*)bp1;
      ap += 4; bp0 += 4; bp1 += 4;
      for (int kk = 4; kk < K; kk += 4) {
        v2f a_n = *(const v2f*)ap;
        v2f b0_n = *(const v2f*)bp0;
        v2f b1_n = *(const v2f*)bp1;
        ap += 4; bp0 += 4; bp1 += 4;
        acc0 = wmma4(a_c, b0_c, acc0);
        acc1 = wmma4(a_c, b1_c, acc1);
        a_c = a_n; b0_c = b0_n; b1_c = b1_n;
      }
      acc0 = wmma4(a_c, b0_c, acc0);
      acc1 = wmma4(a_c, b1_c, acc1);
    } else {
      const float* bp = Bb + (long)(2 * half) * ldb + col0;
      v2f a_c = *(const v2f*)ap;
      v2f b0_c, b1_c;
      b0_c.x = bp[0];  b0_c.y = bp[ldb];
      b1_c.x = bp[16]; b1_c.y = bp[ldb + 16];
      ap += 4; bp += 4L * ldb;
      for (int kk = 4; kk < K; kk += 4) {
        v2f a_n = *(const v2f*)ap;
        v2f b0_n, b1_n;
        b0_n.x = bp[0];  b0_n.y = bp[ldb];
        b1_n.x = bp[16]; b1_n.y = bp[ldb + 16];
        ap += 4; bp += 4L * ldb;
        acc0 = wmma4(a_c, b0_c, acc0);
        acc1 = wmma4(a_c, b1_c, acc1);
        a_c = a_n; b0_c = b0_n; b1_c = b1_n;
      }
      acc0 = wmma4(a_c, b0_c, acc0);
      acc1 = wmma4(a_c, b1_c, acc1);
    }
  } else {
    // single 16-wide subtile (N%32==16 tail), still unpredicated
    if (transB) {
      const float* bp0 = Bb + (long)col0 * ldb + 2 * half;
      for (int kk = 0; kk < K; kk += 4) {
        v2f a = *(const v2f*)ap;
        v2f b = *(const v2f*)bp0;
        acc0 = wmma4(a, b, acc0);
        ap += 4; bp0 += 4;
      }
    } else {
      const float* bp = Bb + (long)(2 * half) * ldb + col0;
      for (int kk = 0; kk < K; kk += 4) {
        v2f a = *(const v2f*)ap;
        v2f b; b.x = bp[0]; b.y = bp[ldb];
        acc0 = wmma4(a, b, acc0);
        ap += 4; bp += 4L * ldb;
      }
    }
  }

  // unguarded epilogue
  float* Cb = C + bz * sC;
  const float* Rb = (Rm != nullptr) ? (Rm + bz * sR) : nullptr;
  {
    const float bv = (bias != nullptr) ? bias[col0] : 0.f;
#pragma unroll
    for (int v = 0; v < 8; ++v) {
      const int m = m0 + v + 8 * half;
      float t = alpha * acc0[v] + bv;
      if (act == 1) t = gelu_f(t);
      if (Rb) t += Rb[(long)m * ldc + col0];
      Cb[(long)m * ldc + col0] = t;
    }
  }
  if (sub1) {
    const int col1 = col0 + 16;
    const float bv = (bias != nullptr) ? bias[col1] : 0.f;
#pragma unroll
    for (int v = 0; v < 8; ++v) {
      const int m = m0 + v + 8 * half;
      float t = alpha * acc1[v] + bv;
      if (act == 1) t = gelu_f(t);
      if (Rb) t += Rb[(long)m * ldc + col1];
      Cb[(long)m * ldc + col1] = t;
    }
  }
}

// ---------------------------------------------------------------------------
// GENERIC guarded GEMM (only for ragged cases: K=3 conv1, M=8 seed FC,
// N=3 output head). 16x16 per wave, per-lane predication.
// ---------------------------------------------------------------------------
__global__ void __launch_bounds__(128)
gemm_f32_wmma_gen(const float* __restrict__ A, const float* __restrict__ Bm,
                  const float* __restrict__ bias, const float* __restrict__ Rm,
                  float* __restrict__ C,
                  int M, int N, int K, int lda, int ldb, int ldc,
                  long sA, long sB, long sC, long sR,
                  int transB, int act, float alpha)
{
  const int wave = threadIdx.x >> 5;
  const int lane = threadIdx.x & 31;
  const int half = lane >> 4;
  const int r    = lane & 15;
  const int m0 = blockIdx.y * 16;
  const int n0 = blockIdx.x * 64 + wave * 16;
  if (n0 >= N) return;
  const long bz = blockIdx.z;
  const float* Ab = A + bz * sA;
  const float* Bb = Bm + bz * sB;
  const int row = m0 + r;
  const int col = n0 + r;
  const bool rOK = row < M;
  const bool cOK = col < N;
  v8f acc = {0.f, 0.f, 0.f, 0.f, 0.f, 0.f, 0.f, 0.f};
  for (int kk = 0; kk < K; kk += 4) {
    const int k0 = kk + 2 * half;
    const int k1 = k0 + 1;
    v2f a, b;
    a.x = (rOK && k0 < K) ? Ab[(long)row * lda + k0] : 0.f;
    a.y = (rOK && k1 < K) ? Ab[(long)row * lda + k1] : 0.f;
    if (transB) {
      b.x = (cOK && k0 < K) ? Bb[(long)col * ldb + k0] : 0.f;
      b.y = (cOK && k1 < K) ? Bb[(long)col * ldb + k1] : 0.f;
    } else {
      b.x = (cOK && k0 < K) ? Bb[(long)k0 * ldb + col] : 0.f;
      b.y = (cOK && k1 < K) ? Bb[(long)k1 * ldb + col] : 0.f;
    }
    acc = wmma4(a, b, acc);
  }
  const float bv = (bias != nullptr && cOK) ? bias[col] : 0.f;
  float* Cb = C + bz * sC;
  const float* Rb = (Rm != nullptr) ? (Rm + bz * sR) : nullptr;
#pragma unroll
  for (int v = 0; v < 8; ++v) {
    const int m = m0 + v + 8 * half;
    if (m < M && cOK) {
      float t = alpha * acc[v] + bv;
      if (act == 1) t = gelu_f(t);
      if (Rb) t += Rb[(long)m * ldc + col];
      Cb[(long)m * ldc + col] = t;
    }
  }
}

// ---------------------------- LayerNorm (last dim) -------------------------
__global__ void __launch_bounds__(128)
ln_kernel(const float* __restrict__ x, const float* __restrict__ g,
          const float* __restrict__ b, float* __restrict__ out, int d)
{
  const long row = blockIdx.x;
  const float* xr = x + row * (long)d;
  float* orow = out + row * (long)d;
  __shared__ float sb[128];
  float s = 0.f;
  for (int i = threadIdx.x; i < d; i += 128) s += xr[i];
  sb[threadIdx.x] = s; __syncthreads();
  for (int st = 64; st > 0; st >>= 1) {
    if (threadIdx.x < st) sb[threadIdx.x] += sb[threadIdx.x + st];
    __syncthreads();
  }
  const float mean = sb[0] / d;
  __syncthreads();
  float s2 = 0.f;
  for (int i = threadIdx.x; i < d; i += 128) { float v = xr[i] - mean; s2 += v * v; }
  sb[threadIdx.x] = s2; __syncthreads();
  for (int st = 64; st > 0; st >>= 1) {
    if (threadIdx.x < st) sb[threadIdx.x] += sb[threadIdx.x + st];
    __syncthreads();
  }
  const float rstd = rsqrtf(sb[0] / d + 1e-5f);
  for (int i = threadIdx.x; i < d; i += 128)
    orow[i] = (xr[i] - mean) * rstd * g[i] + b[i];
}

// ---------------------------- Row softmax ----------------------------------
__global__ void __launch_bounds__(256)
softmax_rows(float* __restrict__ S, int Lk)
{
  const long row = blockIdx.x;
  float* x = S + row * (long)Lk;
  __shared__ float sb[256];
  float mx = -3.4e38f;
  for (int i = threadIdx.x; i < Lk; i += 256) mx = fmaxf(mx, x[i]);
  sb[threadIdx.x] = mx; __syncthreads();
  for (int st = 128; st > 0; st >>= 1) {
    if (threadIdx.x < st) sb[threadIdx.x] = fmaxf(sb[threadIdx.x], sb[threadIdx.x + st]);
    __syncthreads();
  }
  mx = sb[0]; __syncthreads();
  float sm = 0.f;
  for (int i = threadIdx.x; i < Lk; i += 256) { float e = expf(x[i] - mx); x[i] = e; sm += e; }
  sb[threadIdx.x] = sm; __syncthreads();
  for (int st = 128; st > 0; st >>= 1) {
    if (threadIdx.x < st) sb[threadIdx.x] += sb[threadIdx.x + st];
    __syncthreads();
  }
  const float inv = 1.f / sb[0];
  for (int i = threadIdx.x; i < Lk; i += 256) x[i] *= inv;
}

// --------------------- Farthest point sampling (per batch) -----------------
__global__ void __launch_bounds__(256)
fps_kernel(const float* __restrict__ P, int n, int m, int* __restrict__ idx)
{
  const int b = blockIdx.x;
  const int t = threadIdx.x;
  const float* Pb = P + (long)b * n * 3;
  const int cnt = (n + 255) / 256;
  float dlocal[8];
  for (int i = 0; i < 8; ++i) dlocal[i] = 1e10f;
  __shared__ float sval[256];
  __shared__ int   sidx[256];
  __shared__ float lp[3];
  int last = 0;
  for (int it = 0; it < m; ++it) {
    if (t == 0) {
      idx[(long)b * m + it] = last;
      lp[0] = Pb[(long)last * 3 + 0];
      lp[1] = Pb[(long)last * 3 + 1];
      lp[2] = Pb[(long)last * 3 + 2];
    }
    __syncthreads();
    const float bx = lp[0], by = lp[1], bz = lp[2];
    float bestv = -1.f; int besti = 0;
    for (int i = 0; i < cnt; ++i) {
      const int j = t + i * 256;
      if (j < n) {
        const float dx = Pb[(long)j * 3 + 0] - bx;
        const float dy = Pb[(long)j * 3 + 1] - by;
        const float dz = Pb[(long)j * 3 + 2] - bz;
        const float nd = fminf(dlocal[i], dx * dx + dy * dy + dz * dz);
        dlocal[i] = nd;
        if (nd > bestv) { bestv = nd; besti = j; }
      }
    }
    sval[t] = bestv; sidx[t] = besti;
    __syncthreads();
    for (int st = 128; st > 0; st >>= 1) {
      if (t < st) {
        const float v2 = sval[t + st]; const int i2 = sidx[t + st];
        if (v2 > sval[t] || (v2 == sval[t] && i2 < sidx[t])) { sval[t] = v2; sidx[t] = i2; }
      }
      __syncthreads();
    }
    last = sidx[0];
    __syncthreads();
  }
}

// ------------------------------ glue kernels -------------------------------
__global__ void gather_rows(const float* __restrict__ x, const int* __restrict__ idx,
                            float* __restrict__ out, int n, int m, int C, long total)
{
  long i = (long)blockIdx.x * 256 + threadIdx.x;
  if (i >= total) return;
  const long b = i / ((long)m * C);
  const long rrem = i - b * (long)m * C;
  const int mi = (int)(rrem / C);
  const int c = (int)(rrem - (long)mi * C);
  out[i] = x[((long)b * n + idx[(long)b * m + mi]) * C + c];
}

__global__ void concat2(const float* __restrict__ A, const float* __restrict__ B2,
                        float* __restrict__ out, int L, int d1, int d2, long total)
{
  long i = (long)blockIdx.x * 256 + threadIdx.x;
  if (i >= total) return;
  const int D = d1 + d2;
  const long b = i / ((long)L * D);
  const long rrem = i - b * (long)L * D;
  const int l = (int)(rrem / D);
  const int c = (int)(rrem - (long)l * D);
  out[i] = (c < d1) ? A[((long)b * L + l) * d1 + c]
                    : B2[((long)b * L + l) * d2 + (c - d1)];
}

__global__ void pts_transpose(const float* __restrict__ pts, float* __restrict__ P0)
{
  int i = blockIdx.x * 256 + threadIdx.x;               // 8*2048*3
  if (i >= 8 * 2048 * 3) return;
  const int b = i / 6144, rrem = i % 6144;
  const int l = rrem / 3, c = rrem % 3;
  P0[i] = pts[b * 6144 + c * 2048 + l];
}

__global__ void maxpool_tokens(const float* __restrict__ X, float* __restrict__ o1,
                               float* __restrict__ o2, int L, int d)
{
  int i = blockIdx.x * 256 + threadIdx.x;               // 8*d
  if (i >= 8 * d) return;
  const int b = i / d, ch = i % d;
  const float* p = X + (long)b * L * d + ch;
  float m = p[0];
  for (int l = 1; l < L; ++l) m = fmaxf(m, p[(long)l * d]);
  o1[i] = m; o2[i] = m;
}

__global__ void ps_rearrange(const float* __restrict__ flat, const float* __restrict__ psb,
                             float* __restrict__ XX2)
{
  int i = blockIdx.x * 256 + threadIdx.x;               // 8*128*64
  if (i >= 8 * 128 * 64) return;
  const int b = i / 8192, rrem = i % 8192;
  const int t = rrem / 64, o = rrem % 64;
  XX2[i] = gelu_f(flat[b * 8192 + o * 128 + t] + psb[o]);
}

// x2d token-major [B,128,512] -> channel-major reshape [B,256,256] -> token-major Vt
__global__ void vt_reshape(const float* __restrict__ x2d, float* __restrict__ Vt)
{
  long i = (long)blockIdx.x * 256 + threadIdx.x;        // 8*65536
  if (i >= 8L * 65536) return;
  const long b = i >> 16;
  const int rrem = (int)(i & 65535);
  const int l2 = rrem >> 8, c2 = rrem & 255;
  const int f = c2 * 256 + l2;
  const int ch = f >> 7, l = f & 127;
  Vt[i] = x2d[b * 65536 + (long)l * 512 + ch];
}

__global__ void fine_out(const float* __restrict__ fineB, float* __restrict__ out)
{
  int i = blockIdx.x * 256 + threadIdx.x;               // 8*3*256
  if (i >= 8 * 3 * 256) return;
  const int b = i / 768, rrem = i % 768;
  const int ch = rrem / 256, l = rrem % 256;
  out[4096 + i] = fineB[b * 768 + l * 3 + ch];
}

// ------------------------------ host side ----------------------------------
static inline dim3 ew_grid(long n) { return dim3((unsigned)((n + 255) / 256)); }

static void g_gemm(hipStream_t s, const float* A, const float* B, const float* bias,
                   const float* res, float* C, int M, int N, int K,
                   int lda, int ldb, int ldc, long sA, long sB, long sC, long sR,
                   int batch, int transB, int act, float alpha)
{
  const bool fast = ((M & 15) == 0) && ((N & 15) == 0) && ((K & 3) == 0) && (K >= 8);
  if (fast) {
    dim3 grid((N + 127) / 128, M / 16, batch);
    gemm_f32_wmma_fast<<<grid, 128, 0, s>>>(A, B, bias, res, C, M, N, K, lda, ldb, ldc,
                                            sA, sB, sC, sR, transB, act, alpha);
  } else {
    dim3 grid((N + 63) / 64, (M + 15) / 16, batch);
    gemm_f32_wmma_gen<<<grid, 128, 0, s>>>(A, B, bias, res, C, M, N, K, lda, ldb, ldc,
                                           sA, sB, sC, sR, transB, act, alpha);
  }
}

struct CT {
  const float *proj_w, *proj_b, *in_w, *in_b, *out_w, *out_b;
  const float *l1_w, *l1_b, *l2_w, *l2_b, *n12_g, *n12_b, *n13_g, *n13_b;
};

struct Scratch { float *s1, *t1, *q, *mha, *s2, *k, *v, *scores, *h1; };

static void cross_tf(hipStream_t st, const CT& p, int d,
                     const float* src1, int Lq, const float* src2, int Lk,
                     float* out, const Scratch& w)
{
  const int dh = d / HEADS;
  const float scale = 1.f / sqrtf((float)dh);
  const long s1sz = (long)Lq * d, s2sz = (long)Lk * d;
  // proj + pre-LN (n13) on both sources
  g_gemm(st, src1, p.proj_w, p.proj_b, nullptr, w.t1, Lq, d, d, d, d, d, s1sz, 0, s1sz, 0, NB, 1, 0, 1.f);
  ln_kernel<<<NB * Lq, 128, 0, st>>>(w.t1, p.n13_g, p.n13_b, w.s1, d);
  g_gemm(st, src2, p.proj_w, p.proj_b, nullptr, w.k, Lk, d, d, d, d, d, s2sz, 0, s2sz, 0, NB, 1, 0, 1.f);
  ln_kernel<<<NB * Lk, 128, 0, st>>>(w.k, p.n13_g, p.n13_b, w.s2, d);
  // Q/K/V projections
  g_gemm(st, w.s1, p.in_w,               p.in_b,         nullptr, w.q, Lq, d, d, d, d, d, s1sz, 0, s1sz, 0, NB, 1, 0, 1.f);
  g_gemm(st, w.s2, p.in_w + (long)d * d, p.in_b + d,     nullptr, w.k, Lk, d, d, d, d, d, s2sz, 0, s2sz, 0, NB, 1, 0, 1.f);
  g_gemm(st, w.s2, p.in_w + 2L * d * d,  p.in_b + 2 * d, nullptr, w.v, Lk, d, d, d, d, d, s2sz, 0, s2sz, 0, NB, 1, 0, 1.f);
  // attention, per batch (heads batched via z with stride dh)
  for (int b = 0; b < NB; ++b) {
    g_gemm(st, w.q + (long)b * s1sz, w.k + (long)b * s2sz, nullptr, nullptr, w.scores,
           Lq, Lk, dh, d, d, Lk, dh, dh, (long)Lq * Lk, 0, HEADS, 1, 0, scale);
    softmax_rows<<<HEADS * Lq, 256, 0, st>>>(w.scores, Lk);
    g_gemm(st, w.scores, w.v + (long)b * s2sz, nullptr, nullptr, w.mha + (long)b * s1sz,
           Lq, dh, Lk, Lk, d, d, (long)Lq * Lk, dh, dh, 0, HEADS, 0, 0, 1.f);
  }
  // out proj + residual, LN (n12)
  g_gemm(st, w.mha, p.out_w, p.out_b, w.s1, w.t1, Lq, d, d, d, d, d, s1sz, 0, s1sz, s1sz, NB, 1, 0, 1.f);
  ln_kernel<<<NB * Lq, 128, 0, st>>>(w.t1, p.n12_g, p.n12_b, w.s1, d);
  // FFN with residual
  g_gemm(st, w.s1, p.l1_w, p.l1_b, nullptr, w.h1, Lq, FFD, d, d, d, FFD, s1sz, 0, (long)Lq * FFD, 0, NB, 1, 1, 1.f);
  g_gemm(st, w.h1, p.l2_w, p.l2_b, w.s1, out, Lq, d, FFD, FFD, FFD, d, (long)Lq * FFD, 0, s1sz, s1sz, NB, 1, 0, 1.f);
}

extern "C" void kernel_launch(void* const* d_in, const int* in_sizes, int n_in,
                              void* d_out, int out_size, void* d_ws, size_t ws_size,
                              hipStream_t stream)
{
  (void)in_sizes; (void)n_in; (void)out_size; (void)ws_size;
  auto Fp = [&](int i) { return (const float*)d_in[i]; };
  auto getCT = [&](int base) {
    CT c;
    c.proj_w = Fp(base + 0);  c.proj_b = Fp(base + 1);
    c.in_w   = Fp(base + 2);  c.in_b   = Fp(base + 3);
    c.out_w  = Fp(base + 4);  c.out_b  = Fp(base + 5);
    c.l1_w   = Fp(base + 6);  c.l1_b   = Fp(base + 7);
    c.l2_w   = Fp(base + 8);  c.l2_b   = Fp(base + 9);
    c.n12_g  = Fp(base + 10); c.n12_b  = Fp(base + 11);
    c.n13_g  = Fp(base + 12); c.n13_b  = Fp(base + 13);
    return c;
  };
  const float* points   = Fp(0);
  const float* conv1_w  = Fp(1);  const float* conv1_b = Fp(2);
  const float* conv2_w  = Fp(3);  const float* conv2_b = Fp(4);
  const CT sa1   = getCT(5),  sa1_1 = getCT(19), sa2   = getCT(33), sa2_1 = getCT(47);
  const CT sa3   = getCT(61), sa3_1 = getCT(75), sa0_d = getCT(89), sa1_d = getCT(103), sa2_d = getCT(117);
  const float* conv_out_w  = Fp(131); const float* conv_out_b  = Fp(132);
  const float* conv_out1_w = Fp(133); const float* conv_out1_b = Fp(134);
  const float* ps_w        = Fp(135); const float* ps_b        = Fp(136);
  const float* ps_refuse_w = Fp(137); const float* ps_refuse_b = Fp(138);
  const float* ps_adj_w    = Fp(139); const float* ps_adj_b    = Fp(140);
  float* dout = (float*)d_out;

  // ---- workspace carve-up (floats) ----
  float* pool = (float*)d_ws;
  size_t off = 0;
  auto alc = [&](size_t n) { float* p = pool + off; off += n; return p; };
  float* P0 = alc(49152);  float* P1 = alc(12288);  float* P2 = alc(6144);
  float* X1 = alc(1048576); float* X0 = alc(1048576);
  float* Xg0 = alc(262144); float* x1o = alc(262144);
  float* X1cat = alc(524288); float* x1b = alc(524288);
  float* Xg1 = alc(262144); float* x2o = alc(262144);
  float* X2cat = alc(524288); float* x2b = alc(524288);
  float* Xg2 = alc(262144); float* x3o = alc(262144);
  float* X3cat = alc(524288); float* x3b = alc(524288);
  float* xg = alc(4096);  float* xx = alc(4096);
  float* flatb = alc(65536); float* XX2 = alc(65536);
  float* xx3 = alc(524288); float* x0d = alc(524288);
  float* x1d = alc(524288); float* x2d = alc(524288);
  float* Vt = alc(524288);  float* fineA = alc(131072); float* fineB = alc(6144);
  int* idx0 = (int*)alc(4096); int* idx1 = (int*)alc(2048); int* idx2 = (int*)alc(1024);
  Scratch w;
  w.s1 = alc(524288);  w.t1 = alc(524288); w.q = alc(524288); w.mha = alc(524288);
  w.s2 = alc(1048576); w.k = alc(1048576); w.v = alc(1048576);
  w.scores = alc(4194304); w.h1 = alc(4194304);

  // ---- encoder ----
  pts_transpose<<<ew_grid(49152), 256, 0, stream>>>(points, P0);
  g_gemm(stream, P0, conv1_w, conv1_b, nullptr, X1, 2048, 64, 3, 3, 3, 64, 6144, 0, 131072, 0, NB, 1, 1, 1.f);
  g_gemm(stream, X1, conv2_w, conv2_b, nullptr, X0, 2048, 64, 64, 64, 64, 64, 131072, 0, 131072, 0, NB, 1, 0, 1.f);
  fps_kernel<<<NB, 256, 0, stream>>>(P0, 2048, 512, idx0);
  gather_rows<<<ew_grid(8L*512*64), 256, 0, stream>>>(X0, idx0, Xg0, 2048, 512, 64, 8L*512*64);
  gather_rows<<<ew_grid(8L*512*3), 256, 0, stream>>>(P0, idx0, P1, 2048, 512, 3, 8L*512*3);
  cross_tf(stream, sa1, 64, Xg0, 512, X0, 2048, x1o, w);
  concat2<<<ew_grid(8L*512*128), 256, 0, stream>>>(Xg0, x1o, X1cat, 512, 64, 64, 8L*512*128);
  cross_tf(stream, sa1_1, 128, X1cat, 512, X1cat, 512, x1b, w);
  fps_kernel<<<NB, 256, 0, stream>>>(P1, 512, 256, idx1);
  gather_rows<<<ew_grid(8L*256*128), 256, 0, stream>>>(x1b, idx1, Xg1, 512, 256, 128, 8L*256*128);
  gather_rows<<<ew_grid(8L*256*3), 256, 0, stream>>>(P1, idx1, P2, 512, 256, 3, 8L*256*3);
  cross_tf(stream, sa2, 128, Xg1, 256, x1b, 512, x2o, w);
  concat2<<<ew_grid(8L*256*256), 256, 0, stream>>>(Xg1, x2o, X2cat, 256, 128, 128, 8L*256*256);
  cross_tf(stream, sa2_1, 256, X2cat, 256, X2cat, 256, x2b, w);
  fps_kernel<<<NB, 256, 0, stream>>>(P2, 256, 128, idx2);
  gather_rows<<<ew_grid(8L*128*256), 256, 0, stream>>>(x2b, idx2, Xg2, 256, 128, 256, 8L*128*256);
  cross_tf(stream, sa3, 256, Xg2, 128, x2b, 256, x3o, w);
  concat2<<<ew_grid(8L*128*512), 256, 0, stream>>>(Xg2, x3o, X3cat, 128, 256, 256, 8L*128*512);
  cross_tf(stream, sa3_1, 512, X3cat, 128, X3cat, 128, x3b, w);

  // ---- global pooling + seed generator ----
  maxpool_tokens<<<ew_grid(8L*512), 256, 0, stream>>>(x3b, xg, dout, 128, 512);
  g_gemm(stream, xg, ps_adj_w, ps_adj_b, nullptr, xx, 8, 512, 512, 512, 512, 512, 0, 0, 0, 0, 1, 1, 1, 1.f);
  g_gemm(stream, xx, ps_w, nullptr, nullptr, flatb, 8, 8192, 512, 512, 8192, 8192, 0, 0, 0, 0, 1, 0, 0, 1.f);
  ps_rearrange<<<ew_grid(8L*128*64), 256, 0, stream>>>(flatb, ps_b, XX2);
  g_gemm(stream, XX2, ps_refuse_w, ps_refuse_b, nullptr, xx3, 128, 512, 64, 64, 64, 512, 8192, 0, 65536, 0, NB, 1, 1, 1.f);

  // ---- decoder transformers ----
  cross_tf(stream, sa0_d, 512, xx3, 128, xx3, 128, x0d, w);
  cross_tf(stream, sa1_d, 512, x0d, 128, x0d, 128, x1d, w);
  cross_tf(stream, sa2_d, 512, x1d, 128, x1d, 128, x2d, w);

  // ---- output heads ----
  vt_reshape<<<ew_grid(8L*65536), 256, 0, stream>>>(x2d, Vt);
  g_gemm(stream, Vt, conv_out1_w, conv_out1_b, nullptr, fineA, 256, 64, 256, 256, 256, 64, 65536, 0, 16384, 0, NB, 1, 1, 1.f);
  g_gemm(stream, fineA, conv_out_w, conv_out_b, nullptr, fineB, 256, 3, 64, 64, 64, 3, 16384, 0, 768, 0, NB, 1, 0, 1.f);
  fine_out<<<ew_grid(8L*3*256), 256, 0, stream>>>(fineB, dout);
}